// RoDAlignAvg_16320875724972
// MI455X (gfx1250) — compile-verified
//
#include <hip/hip_runtime.h>
#include <stdint.h>

// ---------------------------------------------------------------------------
// RoD-Align average pooling, MI455X (gfx1250).
//
// Phase 1: the 9x9 bilinear sample grid is ROI-independent -> precompute
//          samp[b][c][9*9] (8*256*81 floats = 648 KB) into workspace.
// Phase 2: per ROI: stage the 128-channel half-slab (41472 B) into LDS with
//          global_load_async_to_lds_b128 (ASYNCcnt), compute the per-thread
//          {0, 0.25} exclusion weights in registers, then each output =
//          4 FMAs against LDS, written with non-temporal b128 stores
//          (128 MB write-once stream ~ 5.5 us at 23.3 TB/s).
// ---------------------------------------------------------------------------

typedef float v4f __attribute__((ext_vector_type(4)));

constexpr int B = 8, C = 256, H = 64, W = 64;
constexpr int AH = 9, AW = 9;               // (ALIGNED+1) grid
constexpr int GRID = AH * AW;               // 81
constexpr int SLAB = C * GRID;              // 20736 floats per batch image
constexpr int N_ROIS = 2048;
constexpr int CH = 128;                     // channels per block (half of C)
#define SPATIAL_SCALE 0.0625f

// Matches JAX: bin = (H - 1.001) / (ah - 1.0) computed in double, used as f32.
__device__ __forceinline__ float grid_bin() { return (float)((64.0 - 1.001) / 8.0); }

// ---------------------------------------------------------------------------
// Phase 1: samp[b][c][i*9+j] = bilinear(features[b][c], fixed 9x9 grid)
// ---------------------------------------------------------------------------
__global__ __launch_bounds__(256) void rod_presample(const float* __restrict__ feat,
                                                     float* __restrict__ samp) {
    int idx = blockIdx.x * blockDim.x + threadIdx.x;
    if (idx >= B * SLAB) return;
    int b  = idx / SLAB;
    int r  = idx - b * SLAB;
    int c  = r / GRID;
    int ij = r - c * GRID;
    int i  = ij / AW;
    int j  = ij - i * AW;

    const float bin = grid_bin();
    float h = (float)i * bin;
    float w = (float)j * bin;
    int hs = min((int)floorf(h), H - 2);
    int ws = min((int)floorf(w), W - 2);
    float hr = h - (float)hs;
    float wr = w - (float)ws;

    const float* fb = feat + (size_t)(b * C + c) * (H * W);
    float ul = fb[hs * W + ws];
    float ur = fb[hs * W + ws + 1];
    float dl = fb[(hs + 1) * W + ws];
    float dr = fb[(hs + 1) * W + ws + 1];

    samp[idx] = ul * (1.0f - hr) * (1.0f - wr) + ur * (1.0f - hr) * wr
              + dl * hr * (1.0f - wr)          + dr * hr * wr;
}

// ---------------------------------------------------------------------------
// Phase 2: one block per (ROI, channel-half). 256 threads = 8 waves.
// Thread t owns a fixed 1x4 output tile position (i, j0) and walks channels.
// ---------------------------------------------------------------------------
__global__ __launch_bounds__(256) void rod_pool(const float* __restrict__ samp,
                                                const float* __restrict__ rois,
                                                float* __restrict__ out) {
    __shared__ __align__(16) float s_samp[CH * GRID];   // 41472 B

    const int t  = threadIdx.x;
    const int n  = blockIdx.x;
    const int c0 = blockIdx.y * CH;

    // ROI row (broadcast load; batch index forced uniform -> SGPR base addr)
    const float r0  = rois[n * 5 + 0];
    const float rx1 = rois[n * 5 + 1] * SPATIAL_SCALE;
    const float ry1 = rois[n * 5 + 2] * SPATIAL_SCALE;
    const float rx2 = rois[n * 5 + 3] * SPATIAL_SCALE;
    const float ry2 = rois[n * 5 + 4] * SPATIAL_SCALE;
    const int b = __builtin_amdgcn_readfirstlane((int)r0);

    // ---- async DMA: 128*81 floats = 41472 B = 2592 x b128, global -> LDS ----
    const uint64_t gaddr = (uint64_t)(uintptr_t)(samp + (size_t)b * SLAB + (size_t)c0 * GRID);
    const uint32_t lds_base = (uint32_t)(uintptr_t)(&s_samp[0]);
#pragma unroll
    for (int k = 0; k < 10; ++k) {                      // 10*256 = 2560 x b128
        uint32_t off = (uint32_t)(k * 256 + t) * 16u;
        uint32_t la  = lds_base + off;
        asm volatile("global_load_async_to_lds_b128 %0, %1, %2"
                     :: "v"(la), "v"(off), "s"(gaddr) : "memory");
    }
    if (t < 32) {                                       // remaining 32 x b128
        uint32_t off = (uint32_t)(2560 + t) * 16u;
        uint32_t la  = lds_base + off;
        asm volatile("global_load_async_to_lds_b128 %0, %1, %2"
                     :: "v"(la), "v"(off), "s"(gaddr) : "memory");
    }

    // ---- per-thread exclusion weights, registers only (overlaps the DMA) ----
    // q = (k*256+t)&15 == t&15 -> (i, j0) is k-invariant; only the channel moves.
    const int q  = t & 15;
    const int i  = q >> 1;                              // grid row of 1x4 tile
    const int j0 = (q & 1) * 4;                         // grid col of 1x4 tile
    const int th = t >> 4;                              // channel sub-lane 0..15

    const float bin = grid_bin();
    const float h0 = (float)i * bin;
    const float h1 = (float)(i + 1) * bin;
    const bool rin0 = (h0 >= ry1) && (h0 <= ry2);
    const bool rin1 = (h1 >= ry1) && (h1 <= ry2);
    float m0[5], m1[5];
#pragma unroll
    for (int jj = 0; jj < 5; ++jj) {
        float w = (float)(j0 + jj) * bin;
        bool cin = (w >= rx1) && (w <= rx2);
        m0[jj] = (rin0 && cin) ? 0.0f : 0.25f;          // fold pooling 1/4
        m1[jj] = (rin1 && cin) ? 0.0f : 0.25f;
    }

    asm volatile("s_wait_asynccnt 0x0" ::: "memory");   // my waves' DMAs landed
    __syncthreads();                                    // everyone's landed

    // ---- pooled[c][i][j] = sum_{2x2 corners} mask*samp, 4 outputs/thread ----
    const float* sp = &s_samp[th * GRID + i * AW + j0]; // +1296 floats per k
    float* op = out + ((size_t)n * C + c0 + th) * 64 + i * 8 + j0; // +1024 per k
#pragma unroll
    for (int k = 0; k < 8; ++k) {                       // 2048 float4 / block
        const float* s0 = sp;
        const float* s1 = sp + AW;
        v4f r;
        r.x = m0[0] * s0[0] + m0[1] * s0[1] + m1[0] * s1[0] + m1[1] * s1[1];
        r.y = m0[1] * s0[1] + m0[2] * s0[2] + m1[1] * s1[1] + m1[2] * s1[2];
        r.z = m0[2] * s0[2] + m0[3] * s0[3] + m1[2] * s1[2] + m1[3] * s1[3];
        r.w = m0[3] * s0[3] + m0[4] * s0[4] + m1[3] * s1[3] + m1[4] * s1[4];
        // write-once 128 MB stream: non-temporal b128 store
        __builtin_nontemporal_store(r, (v4f*)op);
        sp += 16 * GRID;                                // next 16 channels
        op += 16 * 64;
    }
}

// ---------------------------------------------------------------------------
extern "C" void kernel_launch(void* const* d_in, const int* in_sizes, int n_in,
                              void* d_out, int out_size, void* d_ws, size_t ws_size,
                              hipStream_t stream) {
    (void)in_sizes; (void)n_in; (void)out_size; (void)ws_size;
    const float* feat = (const float*)d_in[0];   // (8,256,64,64) f32
    const float* rois = (const float*)d_in[1];   // (2048,5) f32
    float* out  = (float*)d_out;                 // (2048,256,8,8) f32
    float* samp = (float*)d_ws;                  // needs 8*256*81*4 = 663552 B

    int total = B * SLAB;                        // 165888 grid samples
    rod_presample<<<(total + 255) / 256, 256, 0, stream>>>(feat, samp);

    dim3 grid(N_ROIS, C / CH);                   // 2048 x 2 blocks
    rod_pool<<<grid, 256, 0, stream>>>(samp, rois, out);
}